// GATLayerAdj_3624952398654
// MI455X (gfx1250) — compile-verified
//
#include <hip/hip_runtime.h>
#include <hip/hip_bf16.h>
#include <math.h>

typedef __attribute__((ext_vector_type(2))) float v2f;
typedef __attribute__((ext_vector_type(8))) float v8f;

#define NN 1024
#define DD 64

// ---------------------------------------------------------------------------
// Kernel 1: s_src'[j,d] = sum_k x[j,k]*F1[d,k] + f_b[d]
//           s_tgt[i,d]  = sum_k x[i,k]*F2[d,k]
// via V_WMMA_F32_16X16X4_F32 (full f32 precision, 16 k-steps of K=4).
// One wave (32 threads) per 16x16 output tile.
//   grid = 64 row-tiles * 4 col-tiles * 2 matrices = 512 blocks of 32.
// A layout (16x4 f32, ISA 7.12.2): lane L, vgpr v -> (M = L%16, K = 2*(L/16)+v)
// B layout (4x16): vgpr v, lane L -> (K = v + 2*(L/16), N = L%16)
// C/D layout (16x16): vgpr r, lane L -> (M = r + 8*(L/16), N = L%16)
// ---------------------------------------------------------------------------
__global__ __launch_bounds__(32)
void gat_proj_wmma_kernel(const float* __restrict__ x,
                          const float* __restrict__ f_w,   // (64,128) row-major
                          const float* __restrict__ f_b,   // (64,)
                          float* __restrict__ s_src,       // (1024,64)
                          float* __restrict__ s_tgt) {     // (1024,64)
  const int bid   = blockIdx.x;
  const int mt    = bid & 63;          // row tile (j / i)
  const int ct    = (bid >> 6) & 3;    // col tile (d)
  const int which = bid >> 8;          // 0 -> F1/s_src, 1 -> F2/s_tgt
  const int lane  = threadIdx.x;
  const int half  = lane >> 4;         // 0|1
  const int l16   = lane & 15;

  const int row  = mt * 16 + l16;      // A-matrix row for this lane
  const int dcol = ct * 16 + l16;      // B/C/D column for this lane

  const float* __restrict__ F = f_w + which * DD;  // F[d,k] = F[d*128 + k]

  v8f c = {};
  #pragma unroll
  for (int kk = 0; kk < 16; ++kk) {
    const int kb = kk * 4 + 2 * half;  // base K for this lane's pair
    v2f a, b;
    a.x = x[row * DD + kb + 0];
    a.y = x[row * DD + kb + 1];
    b.x = F[dcol * 128 + kb + 0];      // B[K][N] = F^T[k+K, dcol] = F[dcol, k+K]
    b.y = F[dcol * 128 + kb + 1];
    c = __builtin_amdgcn_wmma_f32_16x16x4_f32(
        /*neg_a=*/false, a, /*neg_b=*/false, b,
        /*c_mod=*/(short)0, c, /*reuse_a=*/false, /*reuse_b=*/false);
  }

  float* __restrict__ out = (which == 0) ? s_src : s_tgt;
  const float bias = (which == 0) ? f_b[dcol] : 0.0f;
  #pragma unroll
  for (int r = 0; r < 8; ++r) {
    const int m = r + 8 * half;        // output row within tile
    out[(mt * 16 + m) * DD + dcol] = c[r] + bias;
  }
}

// ---------------------------------------------------------------------------
// Kernel 2: u[i] = dot(x[i,:], w_w[0,:64])   (v and w_b cancel in the softmax)
// ---------------------------------------------------------------------------
__global__ __launch_bounds__(256)
void gat_logit_kernel(const float* __restrict__ x,
                      const float* __restrict__ w_w,   // (1,128)
                      float* __restrict__ u) {
  const int i = blockIdx.x * blockDim.x + threadIdx.x;
  if (i >= NN) return;
  float acc = 0.0f;
  #pragma unroll 8
  for (int k = 0; k < DD; ++k) acc = fmaf(x[i * DD + k], w_w[k], acc);
  u[i] = acc;
}

// ---------------------------------------------------------------------------
// Kernel 3: per output row i (one 256-thread block each):
//   M   = max_{j: adj[i,j]>0} u[j]        (row all masked -> uniform weights)
//   w_j = adj[i,j]>0 ? exp(u[j]-M) : 0 ;  Z = sum_j w_j
//   o[i,d] = (1/Z) * sum_j w_j * relu(s_src'[j,d] + s_tgt[i,d])
// Threads: d = t&63 (coalesced over s_src rows), slice = t>>6 (4-way over j).
// ---------------------------------------------------------------------------
__global__ __launch_bounds__(256)
void gat_row_kernel(const int* __restrict__ adj,
                    const float* __restrict__ u,
                    const float* __restrict__ s_src,
                    const float* __restrict__ s_tgt,
                    float* __restrict__ o) {
  __shared__ float lw[NN];
  __shared__ float red[256];
  __shared__ float sM, sZ;

  const int i = blockIdx.x;
  const int t = threadIdx.x;
  const int* __restrict__ arow = adj + (size_t)i * NN;

  // ---- phase 1: row max of u over unmasked entries ----
  float mx = -INFINITY;
  for (int j = t; j < NN; j += 256)
    if (arow[j] > 0) mx = fmaxf(mx, u[j]);
  red[t] = mx;
  __syncthreads();
  for (int s = 128; s > 0; s >>= 1) {
    if (t < s) red[t] = fmaxf(red[t], red[t + s]);
    __syncthreads();
  }
  if (t == 0) sM = red[0];
  __syncthreads();
  const float M = sM;
  const bool uniform = !(M > -INFINITY);   // no edges in this row

  // ---- phase 2: softmax weights into LDS + normalizer ----
  float zs = 0.0f;
  for (int j = t; j < NN; j += 256) {
    float w;
    if (uniform)           w = 1.0f;
    else if (arow[j] > 0)  w = __expf(u[j] - M);
    else                   w = 0.0f;
    lw[j] = w;
    zs += w;
  }
  red[t] = zs;
  __syncthreads();
  for (int s = 128; s > 0; s >>= 1) {
    if (t < s) red[t] += red[t + s];
    __syncthreads();
  }
  if (t == 0) sZ = red[0];
  __syncthreads();
  const float invZ = 1.0f / sZ;
  __syncthreads();   // red[] free for reuse after everyone read sZ

  // ---- phase 3: weighted ReLU accumulation ----
  const int d  = t & 63;
  const int sl = t >> 6;
  const float tv = s_tgt[i * DD + d];
  float acc = 0.0f;
  for (int j = sl; j < NN; j += 4) {
    const float yv = s_src[j * DD + d] + tv;   // f_b already folded into s_src
    acc = fmaf(lw[j], fmaxf(yv, 0.0f), acc);
  }
  red[t] = acc;
  __syncthreads();
  if (sl == 0) {
    const float tot = red[d] + red[d + 64] + red[d + 128] + red[d + 192];
    o[i * DD + d] = tot * invZ;
  }
}

// ---------------------------------------------------------------------------
extern "C" void kernel_launch(void* const* d_in, const int* in_sizes, int n_in,
                              void* d_out, int out_size, void* d_ws, size_t ws_size,
                              hipStream_t stream) {
  const float* x   = (const float*)d_in[0];   // (1024,64)
  const int*   adj = (const int*)  d_in[1];   // (1024,1024)
  // d_in[2..5]: src/tgt/Msrc/Mtgt -- unused by the reference computation
  const float* f_w = (const float*)d_in[6];   // (64,128)
  const float* f_b = (const float*)d_in[7];   // (64,)
  const float* w_w = (const float*)d_in[8];   // (1,128)
  // w_b (d_in[9]) cancels inside the row softmax

  float* out  = (float*)d_out;                // (1024,64)
  float* ssrc = (float*)d_ws;                 // 65536 floats
  float* stgt = ssrc + NN * DD;               // 65536 floats
  float* u    = stgt + NN * DD;               // 1024 floats   (total 516 KB)

  gat_proj_wmma_kernel<<<512, 32, 0, stream>>>(x, f_w, f_b, ssrc, stgt);
  gat_logit_kernel<<<NN / 256, 256, 0, stream>>>(x, w_w, u);
  gat_row_kernel<<<NN, 256, 0, stream>>>(adj, u, ssrc, stgt, out);
}